// MoELoRA_23854248362011
// MI455X (gfx1250) — compile-verified
//
#include <hip/hip_runtime.h>
#include <hip/hip_bf16.h>

typedef __attribute__((ext_vector_type(2))) float v2f;
typedef __attribute__((ext_vector_type(8))) float v8f;

#define TOKENS 8192
#define DDIM   4096
#define HEADS  8
#define RANK   16
#define ODIM   4096
#define CLOW   128   // HEADS*RANK
#define CWIDE  144   // CLOW + 8 router + 8 pad
#define KC     64
#define S1     68    // padded LDS stride (floats), stage 1 (16B-aligned rows)
#define S2     132   // padded LDS stride (floats), stage 2 (16B-aligned rows)

// Async global->LDS copy, 16B per lane, GVS addressing (SGPR base + VGPR byte offset).
__device__ __forceinline__ void async_b128(unsigned lds_byte, const void* base, int goff_bytes) {
  asm volatile("global_load_async_to_lds_b128 %0, %1, %2"
               :: "v"(lds_byte), "v"(goff_bytes), "s"(base) : "memory");
}
__device__ __forceinline__ unsigned lds_addr(const void* p) {
  return (unsigned)(size_t)p;   // low 32 bits of generic ptr = LDS byte offset
}

// ---------------------------------------------------------------------------
// Stage 1: low = x @ [A;Wr]^T (WMMA f32 16x16x4, K=4096), softmax over heads,
//          lowW[token][h*16+r] = low * 16 * softmax_h   (16 = SCALING * H)
// Double-buffered async global->LDS pipeline (ASYNCcnt).
// ---------------------------------------------------------------------------
__global__ __launch_bounds__(256) void moe_lora_stage1(
    const float* __restrict__ x, const float* __restrict__ A,
    const float* __restrict__ Wr, const float* __restrict__ br,
    float* __restrict__ lowW)
{
  __shared__ float Ws[2][CWIDE * S1];   // combined weights [col][k]
  __shared__ float Xs[2][128 * S1];     // x tile           [tok][k]

  const int tid    = threadIdx.x;
  const int lane   = tid & 31;
  const int wave   = tid >> 5;          // 8 waves, 16 tokens each
  const int hl     = lane >> 4;         // half-wave selector
  const int l15    = lane & 15;
  const int tokBlk = blockIdx.x * 128;

  // Zero the padding columns (136..143) once; async staging never touches them.
  #pragma unroll
  for (int b = 0; b < 2; ++b)
    for (int j = tid; j < 8 * KC; j += 256) {
      int col = (CWIDE - 8) + (j >> 6);
      int kk  = j & (KC - 1);
      Ws[b][col * S1 + kk] = 0.0f;
    }

  v8f acc[9];
  #pragma unroll
  for (int t = 0; t < 9; ++t) acc[t] = {};

  // Issue one K-chunk of async copies: exactly 17 per thread (uniform ASYNCcnt).
  auto stage = [&](int buf, int k0) {
    #pragma unroll
    for (int j = 0; j < 8; ++j) {                       // x: 128 tok x 64 k
      int f  = tid + j * 256;
      int tk = f >> 4;
      int k4 = (f & 15) * 4;
      async_b128(lds_addr(&Xs[buf][tk * S1 + k4]), x,
                 ((tokBlk + tk) * DDIM + k0 + k4) * 4);
    }
    #pragma unroll
    for (int j = 0; j < 8; ++j) {                       // A: 128 cols x 64 k
      int f   = tid + j * 256;
      int col = f >> 4;
      int k4  = (f & 15) * 4;
      async_b128(lds_addr(&Ws[buf][col * S1 + k4]), A,
                 (col * DDIM + k0 + k4) * 4);
    }
    {                                                   // Wr: 8 cols x 64 k (dup by upper half)
      int f  = tid & 127;
      int c8 = f >> 4;
      int k4 = (f & 15) * 4;
      async_b128(lds_addr(&Ws[buf][(CLOW + c8) * S1 + k4]), Wr,
                 (c8 * DDIM + k0 + k4) * 4);
    }
  };

  stage(0, 0);
  const int NCH = DDIM / KC;
  for (int i = 0; i < NCH; ++i) {
    const int buf = i & 1;
    if (i + 1 < NCH) {
      stage(buf ^ 1, (i + 1) * KC);
      asm volatile("s_wait_asynccnt 0x11" ::: "memory");   // chunk i retired (17 in flight)
    } else {
      asm volatile("s_wait_asynccnt 0x0" ::: "memory");
    }
    __syncthreads();

    const float* xrow = &Xs[buf][(wave * 16 + l15) * S1];
    const float* wcol = &Ws[buf][l15 * S1];
    #pragma unroll 4
    for (int kk = 0; kk < KC; kk += 4) {
      const int kb = kk + hl * 2;                 // lanes 16-31 take K=2,3
      v2f a = *(const v2f*)(xrow + kb);           // ds_load_b64
      #pragma unroll
      for (int t = 0; t < 9; ++t) {
        v2f b = *(const v2f*)(wcol + t * 16 * S1 + kb);
        acc[t] = __builtin_amdgcn_wmma_f32_16x16x4_f32(
            false, a, false, b, (short)0, acc[t], false, false);
      }
    }
    __syncthreads();   // buffer fully consumed before it is refilled
  }

  // Epilogue: tile 8 cols 0..7 hold router logits (head = lane N).
  float wv[8];
  const float bias = br[l15 & 7];
  #pragma unroll
  for (int i = 0; i < 8; ++i) {
    float l = acc[8][i] + bias;
    float m = l;
    m = fmaxf(m, __shfl_xor(m, 1, 8));
    m = fmaxf(m, __shfl_xor(m, 2, 8));
    m = fmaxf(m, __shfl_xor(m, 4, 8));
    float e = __expf(l - m);
    float s = e;
    s += __shfl_xor(s, 1, 8);
    s += __shfl_xor(s, 2, 8);
    s += __shfl_xor(s, 4, 8);
    wv[i] = 16.0f * e / s;    // SCALING(2) * H(8) * softmax
  }
  #pragma unroll
  for (int t = 0; t < 8; ++t) {
    #pragma unroll
    for (int i = 0; i < 8; ++i) {
      float sc = __shfl(wv[i], t, 16);           // head-t weight per half-wave
      int token = tokBlk + wave * 16 + i + hl * 8;
      lowW[token * CLOW + t * 16 + l15] = acc[t][i] * sc;
    }
  }
}

// ---------------------------------------------------------------------------
// Stage 2: out[token][o] = sum_{h,r} lowW[token][h*16+r] * B[h][o][r]
//          GEMM [8192 x 128] x [128 x 4096], WMMA f32 16x16x4.
//          One-shot async panel staging (full K=128 fits in LDS).
// ---------------------------------------------------------------------------
__global__ __launch_bounds__(256) void moe_lora_stage2(
    const float* __restrict__ lowW, const float* __restrict__ Bm,
    float* __restrict__ out)
{
  __shared__ float Xl[128 * S2];   // lowW tile [tok][k]  (full K=128)
  __shared__ float Wl[64 * S2];    // B slice   [o][k], k = h*16+r

  const int tid    = threadIdx.x;
  const int lane   = tid & 31;
  const int wave   = tid >> 5;
  const int hl     = lane >> 4;
  const int l15    = lane & 15;
  const int o0     = blockIdx.x * 64;
  const int tokBlk = blockIdx.y * 128;

  #pragma unroll
  for (int j = 0; j < 16; ++j) {                 // lowW: 128 tok x 128 k
    int f  = tid + j * 256;
    int tk = f >> 5;
    int k4 = (f & 31) * 4;
    async_b128(lds_addr(&Xl[tk * S2 + k4]), lowW,
               ((tokBlk + tk) * CLOW + k4) * 4);
  }
  #pragma unroll
  for (int j = 0; j < 8; ++j) {                  // B: [h][o][r] -> Wl[o][h*16+r]
    int f  = tid + j * 256;
    int h  = f >> 8;
    int o  = (f >> 2) & 63;
    int r4 = (f & 3) * 4;
    async_b128(lds_addr(&Wl[o * S2 + h * 16 + r4]), Bm,
               (h * (ODIM * RANK) + (o0 + o) * RANK + r4) * 4);
  }
  asm volatile("s_wait_asynccnt 0x0" ::: "memory");
  __syncthreads();

  v8f acc[4];
  #pragma unroll
  for (int c = 0; c < 4; ++c) acc[c] = {};

  const float* xrow = &Xl[(wave * 16 + l15) * S2];
  const float* wrow = &Wl[l15 * S2];
  #pragma unroll 4
  for (int kk = 0; kk < CLOW; kk += 4) {
    const int kb = kk + hl * 2;
    v2f a = *(const v2f*)(xrow + kb);
    #pragma unroll
    for (int c = 0; c < 4; ++c) {
      v2f b = *(const v2f*)(wrow + c * 16 * S2 + kb);
      acc[c] = __builtin_amdgcn_wmma_f32_16x16x4_f32(
          false, a, false, b, (short)0, acc[c], false, false);
    }
  }

  #pragma unroll
  for (int c = 0; c < 4; ++c) {
    #pragma unroll
    for (int i = 0; i < 8; ++i) {
      int token = tokBlk + wave * 16 + i + hl * 8;
      out[token * ODIM + o0 + c * 16 + l15] = acc[c][i];
    }
  }
}

// ---------------------------------------------------------------------------
extern "C" void kernel_launch(void* const* d_in, const int* in_sizes, int n_in,
                              void* d_out, int out_size, void* d_ws, size_t ws_size,
                              hipStream_t stream) {
  const float* x  = (const float*)d_in[0];   // [4,2048,4096]
  const float* A  = (const float*)d_in[1];   // [8,16,4096]
  const float* Bm = (const float*)d_in[2];   // [8,4096,16]
  const float* Wr = (const float*)d_in[3];   // [8,4096]
  const float* br = (const float*)d_in[4];   // [8]
  float* out  = (float*)d_out;               // [4,2048,4096]
  float* lowW = (float*)d_ws;                // [8192,128] scratch (4 MB)

  moe_lora_stage1<<<TOKENS / 128, 256, 0, stream>>>(x, A, Wr, br, lowW);
  moe_lora_stage2<<<dim3(ODIM / 64, TOKENS / 128), 256, 0, stream>>>(lowW, Bm, out);
}